// ARIMAModel_90263032693081
// MI455X (gfx1250) — compile-verified
//
#include <hip/hip_runtime.h>
#include <stdint.h>

typedef __attribute__((ext_vector_type(2))) float v2f;
typedef __attribute__((ext_vector_type(8))) float v8f;

// Problem constants (ARIMA p=2, d=1, q=2 on (1024, 65536) f32)
constexpr int BROWS   = 1024;
constexpr int LEN     = 65536;
constexpr int TOUT    = LEN - 1;       // 65535 output columns
constexpr int TVALID  = TOUT - 2;      // 65533 valid eps, last 2 exact zeros
constexpr int TILE    = 16;            // 16x16 WMMA output tile
constexpr int BLOCK_T = 256;           // 16 tiles/block, 2 tiles per wave
constexpr int TPW     = 2;             // tiles per wave
constexpr int NKBLK   = 12;            // 48-tap window -> 12 x K=4 WMMA steps
constexpr int CW      = BLOCK_T + 32;  // 288 c values per block window
constexpr int YW      = 292;           // y window floats (mult of 4, covers CW+2)
constexpr int YSTR    = 296;           // padded LDS row stride (1184 B, 16B-mult)
constexpr int CSTR    = 292;           // c LDS row stride
constexpr int HOFF    = 16;            // h table holds indices -16..47
constexpr int HN      = 64;

__global__ __launch_bounds__(256)
void arima_fir_wmma_kernel(const float* __restrict__ y,
                           const float* __restrict__ phi,
                           const float* __restrict__ theta,
                           const float* __restrict__ mu,
                           float* __restrict__ out)
{
    __shared__ float ly[16 * YSTR];  // staged y window (async-to-LDS)
    __shared__ float lc[16 * CSTR];  // c_t window shared by all 16 tiles
    __shared__ float lh[HN];         // FIR taps h_j, j = -16..47 (zeros j<0)

    const int tid  = threadIdx.x;
    const int lane = tid & 31;
    const int wave = tid >> 5;
    const int tc0  = blockIdx.x * BLOCK_T;  // output-time base of this block
    const int rb   = blockIdx.y * TILE;     // batch-row base of this block

    const float phi0 = phi[0], phi1 = phi[1];
    const float th0  = theta[0], th1 = theta[1];
    const float m    = mu[0];

    // --- impulse response of 1/(1 + th0 z^-1 + th1 z^-2), 48 taps ----------
    if (tid == 0) {
        #pragma unroll
        for (int j = 0; j < HOFF; ++j) lh[j] = 0.0f;
        float hm2 = 0.0f, hm1 = 1.0f;
        lh[HOFF + 0] = 1.0f;
        #pragma unroll
        for (int j = 1; j < 48; ++j) {
            float h = -th0 * hm1 - th1 * hm2;
            lh[HOFF + j] = h;
            hm2 = hm1; hm1 = h;
        }
    }

    // --- stage y window [g0, g0+YW) into LDS via async b128 copies ---------
    // g0 = tc0-32 is 16B-aligned; interior chunks use GLOBAL_LOAD_ASYNC_TO_LDS,
    // boundary chunks (first/last block only) fall back to guarded stores.
    const int r  = tid >> 4;   // row 0..15
    const int cb = tid & 15;   // column slice within row (coalesced)
    const int g0 = tc0 - 32;
    const float* yblk = y + (size_t)rb * LEN;                 // uniform base
    const uint32_t lyb = (uint32_t)(uintptr_t)(&ly[0]);       // LDS byte base

    #pragma unroll
    for (int k = 0; k < 5; ++k) {
        const int q = cb + 16 * k;                 // float4 chunk index
        if (q < YW / 4) {
            const int col = 4 * q;
            const int g   = g0 + col;
            if (g >= 0 && g + 3 < LEN) {
                const uint32_t ldsaddr = lyb + (uint32_t)(r * YSTR + col) * 4u;
                const int      voff    = (r * LEN + g) * 4;   // byte offset
                asm volatile("global_load_async_to_lds_b128 %0, %1, %2 offset:0"
                             :: "v"(ldsaddr), "v"(voff), "s"(yblk) : "memory");
            } else {
                #pragma unroll
                for (int e = 0; e < 4; ++e) {
                    const int ge = g + e;
                    ly[r * YSTR + col + e] =
                        (ge >= 0 && ge < LEN) ? yblk[(size_t)r * LEN + ge] : 0.0f;
                }
            }
        }
    }
    asm volatile("s_wait_asynccnt 0x0" ::: "memory");
    __syncthreads();

    // --- c_t = (y[t+3]-y[t+2]) - mu - phi0*y[t+2] - phi1*y[t+1] ------------
    // c col j <-> t = tc0-32+j ; y[t+1] sits at ly col j+1 (col0 <-> g0).
    for (int j = cb; j < CW; j += 16) {
        const int t = tc0 - 32 + j;
        float v = 0.0f;
        if (t >= 0 && t < TVALID) {
            const float* yr = &ly[r * YSTR];
            const float y1 = yr[j + 1], y2 = yr[j + 2], y3 = yr[j + 3];
            v = (y3 - y2) - m - (phi0 * y2 + phi1 * y1);
        }
        lc[r * CSTR + j] = v;
    }
    __syncthreads();

    // --- per-lane fragment coordinates --------------------------------------
    const int n  = lane & 15;   // column-in-tile (B/C/D N index, A M index)
    const int hi = lane >> 4;   // lane half selects K pair / row group

    // B fragments: Bm[k][n] = h_{32 + n - (4*blk + k)}; reused by both tiles.
    v2f bfrag[NKBLK];
    #pragma unroll
    for (int blk = 0; blk < NKBLK; ++blk) {
        const int kb = 4 * blk + 2 * hi;
        bfrag[blk].x = lh[HOFF + 32 + n - kb];
        bfrag[blk].y = lh[HOFF + 32 + n - kb - 1];
    }

    // --- two 16x16 tiles per wave: 2 x 12 chained f32 WMMAs ----------------
    #pragma unroll
    for (int tt = 0; tt < TPW; ++tt) {
        const int tile = wave * TPW + tt;       // 0..15 across block
        v8f acc = {};
        #pragma unroll
        for (int blk = 0; blk < NKBLK; ++blk) {
            const int j = tile * TILE + 4 * blk + 2 * hi;
            v2f a;
            a.x = lc[n * CSTR + j];
            a.y = lc[n * CSTR + j + 1];
            acc = __builtin_amdgcn_wmma_f32_16x16x4_f32(
                /*neg_a=*/false, a, /*neg_b=*/false, bfrag[blk],
                /*c_mod=*/(short)0, acc, /*reuse_a=*/false, /*reuse_b=*/false);
        }

        // store D (f32 C/D layout: VGPR v -> rows v and v+8)
        const int ocol = tc0 + tile * TILE + n;
        if (ocol < TOUT) {
            const bool valid = ocol < TVALID;   // reference appends exact zeros
            #pragma unroll
            for (int v = 0; v < 8; ++v) {
                const int orow = rb + v + 8 * hi;
                out[(size_t)orow * TOUT + ocol] = valid ? acc[v] : 0.0f;
            }
        }
    }
}

extern "C" void kernel_launch(void* const* d_in, const int* in_sizes, int n_in,
                              void* d_out, int out_size, void* d_ws, size_t ws_size,
                              hipStream_t stream) {
    const float* y     = (const float*)d_in[0];
    const float* phi   = (const float*)d_in[1];
    const float* theta = (const float*)d_in[2];
    const float* mu    = (const float*)d_in[3];
    float* out = (float*)d_out;

    dim3 grid(LEN / BLOCK_T, BROWS / TILE);  // (256, 64)
    arima_fir_wmma_kernel<<<grid, dim3(256), 0, stream>>>(y, phi, theta, mu, out);
}